// Group_2903397892248
// MI455X (gfx1250) — compile-verified
//
#include <hip/hip_runtime.h>

typedef float v2f __attribute__((ext_vector_type(2)));
typedef float v8f __attribute__((ext_vector_type(8)));

constexpr int B_   = 8;
constexpr int N_   = 16384;
constexpr int G_   = 512;   // num_group
constexpr int M_   = 32;    // group_size
constexpr int GT   = 16;    // centers (rows) per KNN block = WMMA M
constexpr int CHUNK = 1024; // columns per LDS chunk
constexpr float BIGF = 1e10f;

// ---------------------------------------------------------------------------
// Kernel 1: farthest point sampling. One block per batch, 1024 threads,
// 16 points per thread held in VGPRs. Writes centers[b][g][3].
// ---------------------------------------------------------------------------
__global__ __launch_bounds__(1024) void fps_kernel(const float* __restrict__ xyz,
                                                   float* __restrict__ centers) {
  const int b = blockIdx.x;
  const int t = threadIdx.x;
  const float* pts = xyz + (size_t)b * N_ * 3;

  float px[16], py[16], pz[16], dd[16];
#pragma unroll
  for (int s = 0; s < 16; ++s) {
    const int p = s * 1024 + t;
    px[s] = pts[p * 3 + 0];
    py[s] = pts[p * 3 + 1];
    pz[s] = pts[p * 3 + 2];
    dd[s] = BIGF;
  }

  __shared__ float s_val[32];
  __shared__ int   s_idx[32];
  __shared__ float s_c[3];
  __shared__ int   s_cur;

  const int wave = t >> 5;
  const int lane = t & 31;
  int cur = 0;

  for (int k = 0; k < G_; ++k) {
    // broadcast centroid = xyz[cur]
    if (t == (cur & 1023)) {
      const int s = cur >> 10;
      s_c[0] = px[s]; s_c[1] = py[s]; s_c[2] = pz[s];
    }
    __syncthreads();
    const float cx = s_c[0], cy = s_c[1], cz = s_c[2];
    if (t == 0) {
      float* co = centers + ((size_t)b * G_ + k) * 3;
      co[0] = cx; co[1] = cy; co[2] = cz;
    }

    // dist update + local argmax (ties -> smallest index)
    float bv = -1.0f;
    int   bi = 0x7FFFFFFF;
#pragma unroll
    for (int s = 0; s < 16; ++s) {
      const float dx = px[s] - cx, dy = py[s] - cy, dz = pz[s] - cz;
      float d = dx * dx + dy * dy + dz * dz;
      d = fminf(dd[s], d);
      dd[s] = d;
      const int gi = s * 1024 + t;
      if (d > bv || (d == bv && gi < bi)) { bv = d; bi = gi; }
    }
    // wave32 reduce
#pragma unroll
    for (int m = 1; m < 32; m <<= 1) {
      const float ov = __shfl_xor(bv, m);
      const int   oi = __shfl_xor(bi, m);
      if (ov > bv || (ov == bv && oi < bi)) { bv = ov; bi = oi; }
    }
    if (lane == 0) { s_val[wave] = bv; s_idx[wave] = bi; }
    __syncthreads();
    if (wave == 0) {
      bv = s_val[lane];
      bi = s_idx[lane];
#pragma unroll
      for (int m = 1; m < 32; m <<= 1) {
        const float ov = __shfl_xor(bv, m);
        const int   oi = __shfl_xor(bi, m);
        if (ov > bv || (ov == bv && oi < bi)) { bv = ov; bi = oi; }
      }
      if (lane == 0) s_cur = bi;
    }
    __syncthreads();
    cur = s_cur;
  }
}

// ---------------------------------------------------------------------------
// Kernel 2: KNN (top-32 by squared distance) + gather + recentre.
// One block per 16 centers. Distances d = |c|^2 + |x|^2 - 2 c.x computed in
// 16x16 tiles with V_WMMA_F32_16X16X4_F32:
//   A[r] = (-2cx, -2cy, -2cz, |c|^2),  B[:,n] = (x, y, z, 1)  =>  |c|^2 - 2c.x
// then each lane adds |x|^2 for its column. Streaming top-32 per row via
// masked argmin passes over (chunk in LDS) U (carry list).
// ---------------------------------------------------------------------------
__global__ __launch_bounds__(256) void knn_gather_kernel(const float* __restrict__ xyz,
                                                         const float* __restrict__ centers,
                                                         float* __restrict__ nbh) {
  __shared__ float sD[GT * CHUNK];       // 64 KB distance chunk (row-major)
  __shared__ float sCarryD[2][GT][M_];   // 4 KB running top-32 dists
  __shared__ int   sCarryI[2][GT][M_];   // 4 KB running top-32 indices

  const int b    = blockIdx.x / (G_ / GT);
  const int g0   = (blockIdx.x % (G_ / GT)) * GT;
  const int t    = threadIdx.x;
  const int wave = t >> 5;   // 0..7
  const int lane = t & 31;
  const int half = lane >> 4;
  const int l16  = lane & 15;

  const float* pts = xyz + (size_t)b * N_ * 3;

  // A matrix (constant per block). 16x4 f32 layout: lanes 0-15 hold row=l16
  // K0,K1; lanes 16-31 hold row=l16 K2,K3.
  float cr0, cr1, cr2;
  {
    const float* c = centers + ((size_t)b * G_ + g0 + l16) * 3;
    cr0 = c[0]; cr1 = c[1]; cr2 = c[2];
  }
  const float cc = cr0 * cr0 + cr1 * cr1 + cr2 * cr2;
  v2f a;
  a.x = (half == 0) ? (-2.0f * cr0) : (-2.0f * cr2);
  a.y = (half == 0) ? (-2.0f * cr1) : cc;

  // init carry buffer 0
  for (int e = t; e < GT * M_; e += 256) {
    sCarryD[0][e / M_][e % M_] = BIGF;
    sCarryI[0][e / M_][e % M_] = 0x7FFFFFFF;
  }
  __syncthreads();

  const int row_g = t >> 4;  // selection group's row, 0..15
  const int gl    = t & 15;  // lane within 16-thread group (half-wave)
  int buf = 0;

  for (int c0 = 0; c0 < N_; c0 += CHUNK) {
    // ---- compute distance tile into LDS via WMMA (8 waves x 8 subtiles) ----
    for (int s = wave; s < CHUNK / 16; s += 8) {
      const int col = c0 + s * 16 + l16;
      const float x = pts[col * 3 + 0];
      const float y = pts[col * 3 + 1];
      const float z = pts[col * 3 + 2];
      const float xx = x * x + y * y + z * z;
      v2f bb;
      bb.x = (half == 0) ? x : z;
      bb.y = (half == 0) ? y : 1.0f;
      v8f acc = {};
      acc = __builtin_amdgcn_wmma_f32_16x16x4_f32(
          /*neg_a=*/false, a, /*neg_b=*/false, bb,
          /*c_mod=*/(short)0, acc, /*reuse_a=*/false, /*reuse_b=*/false);
      const int colLocal = s * 16 + l16;
      const int m0 = half * 8;  // C/D rows: lanes<16 -> M=0..7, lanes>=16 -> M=8..15
#pragma unroll
      for (int i = 0; i < 8; ++i)
        sD[(m0 + i) * CHUNK + colLocal] = acc[i] + xx;
    }
    __syncthreads();

    // ---- merge: top-32 of (chunk row U old carry) -> new carry, sorted ----
    const int nb = buf ^ 1;
    for (int k = 0; k < M_; ++k) {
      float bvd = 4.0f * BIGF;
      int   bvi = 0x7FFFFFFF;
      int   bloc = 0;
      for (int cl = gl; cl < CHUNK; cl += 16) {
        const float d = sD[row_g * CHUNK + cl];
        const int  gi = c0 + cl;
        if (d < bvd || (d == bvd && gi < bvi)) { bvd = d; bvi = gi; bloc = cl; }
      }
#pragma unroll
      for (int e = gl; e < M_; e += 16) {
        const float d = sCarryD[buf][row_g][e];
        const int  gi = sCarryI[buf][row_g][e];
        if (d < bvd || (d == bvd && gi < bvi)) { bvd = d; bvi = gi; bloc = CHUNK + e; }
      }
      // half-wave (16 lane) argmin reduce, ties -> smaller global index
#pragma unroll
      for (int m = 1; m < 16; m <<= 1) {
        const float od = __shfl_xor(bvd, m);
        const int   oi = __shfl_xor(bvi, m);
        const int   ol = __shfl_xor(bloc, m);
        if (od < bvd || (od == bvd && oi < bvi)) { bvd = od; bvi = oi; bloc = ol; }
      }
      if (gl == 0) {
        sCarryD[nb][row_g][k] = bvd;
        sCarryI[nb][row_g][k] = bvi;
        if (bloc >= CHUNK) sCarryD[buf][row_g][bloc - CHUNK] = 8.0f * BIGF;
        else               sD[row_g * CHUNK + bloc]          = 8.0f * BIGF;
      }
      __syncthreads();
    }
    buf = nb;
    __syncthreads();
  }

  // ---- output: neighborhood[b][g0+row][k] = xyz[idx] - center ----
  {
    const float* c = centers + ((size_t)b * G_ + g0 + row_g) * 3;
    const float cox = c[0], coy = c[1], coz = c[2];
    for (int k = gl; k < M_; k += 16) {
      const int gi = sCarryI[buf][row_g][k];
      const float x = pts[gi * 3 + 0];
      const float y = pts[gi * 3 + 1];
      const float z = pts[gi * 3 + 2];
      float* o = nbh + (((size_t)b * G_ + g0 + row_g) * M_ + k) * 3;
      o[0] = x - cox; o[1] = y - coy; o[2] = z - coz;
    }
  }
}

// ---------------------------------------------------------------------------
extern "C" void kernel_launch(void* const* d_in, const int* in_sizes, int n_in,
                              void* d_out, int out_size, void* d_ws, size_t ws_size,
                              hipStream_t stream) {
  (void)in_sizes; (void)n_in; (void)out_size; (void)d_ws; (void)ws_size;
  const float* xyz = (const float*)d_in[0];
  float* out = (float*)d_out;
  float* nbh     = out;                             // [B][G][M][3]
  float* centers = out + (size_t)B_ * G_ * M_ * 3;  // [B][G][3]

  fps_kernel<<<B_, 1024, 0, stream>>>(xyz, centers);
  knn_gather_kernel<<<B_ * (G_ / GT), 256, 0, stream>>>(xyz, centers, nbh);
}